// SAGENet_59150289601024
// MI455X (gfx1250) — compile-verified
//
#include <hip/hip_runtime.h>
#include <hip/hip_bf16.h>

// ---------------------------------------------------------------------------
// GraphSAGE on gfx1250.
//  - Edge phase: coalesced scatter-add atomics (roofline-dominant, ~1.8 GB).
//  - Dense phase: f32 WMMA 16x16x4; pre-padded A buffers, weights pre-packed
//    into wave32 fragment order (unguarded coalesced b64 loads), K fully
//    unrolled, A hoisted across N tiles, dual accumulators to break the
//    WMMA RAW chain, compile-time RELU.
// ---------------------------------------------------------------------------

typedef __attribute__((ext_vector_type(2))) float v2f;
typedef __attribute__((ext_vector_type(8))) float v8f;

#if __has_builtin(__builtin_amdgcn_wmma_f32_16x16x4_f32)
#define HAVE_WMMA_F32X4 1
#endif

__device__ inline v8f wmma_f32x4(v2f a, v2f b, v8f c) {
#ifdef HAVE_WMMA_F32X4
    return __builtin_amdgcn_wmma_f32_16x16x4_f32(
        false, a, false, b, (short)0, c, false, false);
#else
    const int lane  = threadIdx.x & 31;
    const int mlo   = lane & 15;
    const int khalf = lane >> 4;
    v8f out = c;
    #pragma unroll
    for (int k = 0; k < 4; ++k) {
        const int srcGrp = (k >> 1) * 16;
        const float bv = __shfl((k & 1) ? b[1] : b[0], srcGrp + mlo, 32);
        #pragma unroll
        for (int v = 0; v < 8; ++v) {
            const int m = v + 8 * khalf;
            const float av = __shfl((k & 1) ? a[1] : a[0], srcGrp + m, 32);
            out[v] += av * bv;
        }
    }
    return out;
#endif
}

// ---------------------------------------------------------------------------
// Prep kernels
// ---------------------------------------------------------------------------

__global__ void pad_rows_kernel(const float* __restrict__ X, float* __restrict__ Xp,
                                int N, int K, int KPAD) {
    long long idx = (long long)blockIdx.x * blockDim.x + threadIdx.x;
    if (idx >= (long long)N * KPAD) return;
    int r = (int)(idx / KPAD);
    int k = (int)(idx - (long long)r * KPAD);
    Xp[idx] = (k < K) ? X[(size_t)r * K + k] : 0.0f;
}

// Pack W[K][Nout] into WMMA B-fragment order: Wf[((nt*kSteps+ks)*32+lane)*2+{0,1}]
// lane: n = nt*16 + (lane&15), k = ks*4 + (lane>>4)*2 + {0,1}; zero-fill OOB.
__global__ void pack_weights_kernel(const float* __restrict__ W, const float* __restrict__ bias,
                                    int K, int Nout, int nTiles, int kSteps,
                                    float* __restrict__ Wf, float* __restrict__ biasPad) {
    int idx = blockIdx.x * blockDim.x + threadIdx.x;
    int total = nTiles * kSteps * 32;
    if (idx < total) {
        int lane = idx & 31;
        int ks = (idx >> 5) % kSteps;
        int nt = (idx >> 5) / kSteps;
        int n = nt * 16 + (lane & 15);
        int kb = ks * 4 + (lane >> 4) * 2;
        float v0 = (kb     < K && n < Nout) ? W[(size_t)kb       * Nout + n] : 0.0f;
        float v1 = (kb + 1 < K && n < Nout) ? W[(size_t)(kb + 1) * Nout + n] : 0.0f;
        Wf[(size_t)idx * 2    ] = v0;
        Wf[(size_t)idx * 2 + 1] = v1;
    }
    if (idx < nTiles * 16) {
        biasPad[idx] = (idx < Nout) ? bias[idx] : 0.0f;
    }
}

// ---------------------------------------------------------------------------
// Edge-phase kernels
// ---------------------------------------------------------------------------

__global__ void degree_kernel(const int* __restrict__ dst, float* __restrict__ cnt, int E) {
    int e = blockIdx.x * blockDim.x + threadIdx.x;
    if (e < E) atomicAdd(&cnt[dst[e]], 1.0f);
}

__global__ void rcnt_kernel(float* __restrict__ cnt, int N) {
    int i = blockIdx.x * blockDim.x + threadIdx.x;
    if (i < N) cnt[i] = 1.0f / fmaxf(cnt[i], 1.0f);
}

__global__ void scatter_sum_kernel(const float* __restrict__ X,
                                   const int* __restrict__ src,
                                   const int* __restrict__ dst,
                                   float* __restrict__ S, int E, int F, int XS, int SS) {
    long long idx = (long long)blockIdx.x * blockDim.x + threadIdx.x;
    if (idx >= (long long)E * F) return;
    int e = (int)(idx / F);
    int f = (int)(idx - (long long)e * F);
    int s = src[e];
    int d = dst[e];
    atomicAdd(&S[(size_t)d * SS + f], X[(size_t)s * XS + f]);
}

// ---------------------------------------------------------------------------
// Fused linear: out = act( rcnt*A1 @ W1 + bias [+ A2 @ W2] )
// A row stride = KPAD (multiple of 4, zero-padded). Weights in fragment
// order. Output columns < Nstore, row stride Nstore. M multiple of 16.
// Two accumulator chains to break WMMA RAW dependencies.
// ---------------------------------------------------------------------------

template <int KPAD, bool DUAL, bool RELU>
__global__ __launch_bounds__(128)
void fused_linear_wmma(const float* __restrict__ A1, const float* __restrict__ rcnt,
                       const float* __restrict__ W1f,
                       const float* __restrict__ A2, const float* __restrict__ W2f,
                       const float* __restrict__ biasPad, float* __restrict__ out,
                       int M, int nTiles, int Nstore) {
    constexpr int KS = KPAD / 4;
    const int lane  = threadIdx.x & 31;
    const int warp  = threadIdx.x >> 5;
    const int mTile = blockIdx.x * 4 + warp;
    if (mTile * 16 >= M) return;               // wave-uniform
    const int rowBase = mTile * 16;
    const int mlo   = lane & 15;
    const int khalf = lane >> 4;

    float rs = 1.0f;
    if (rcnt) rs = rcnt[rowBase + mlo];

    // A fragments: load once, reuse across all N tiles (unguarded b64 loads).
    v2f a1[KS];
    v2f a2[KS];
    {
        const float* p = A1 + (size_t)(rowBase + mlo) * KPAD + khalf * 2;
        #pragma unroll
        for (int ks = 0; ks < KS; ++ks) {
            v2f t = *(const v2f*)(p + ks * 4);
            t[0] *= rs; t[1] *= rs;
            a1[ks] = t;
        }
        if (DUAL) {
            const float* q = A2 + (size_t)(rowBase + mlo) * KPAD + khalf * 2;
            #pragma unroll
            for (int ks = 0; ks < KS; ++ks) a2[ks] = *(const v2f*)(q + ks * 4);
        }
    }

    for (int nt = 0; nt < nTiles; ++nt) {
        const int ncol = nt * 16 + mlo;
        const float bval = biasPad[ncol];
        // Two independent accumulator chains.
        v8f accA, accB;
        #pragma unroll
        for (int i = 0; i < 8; ++i) { accA[i] = bval; accB[i] = 0.0f; }

        const v2f* w1 = (const v2f*)W1f + (size_t)nt * KS * 32 + lane;
        const v2f* w2 = DUAL ? ((const v2f*)W2f + (size_t)nt * KS * 32 + lane) : nullptr;

        if (DUAL) {
            #pragma unroll
            for (int ks = 0; ks < KS; ++ks) {
                v2f bf = w1[(size_t)ks * 32];      // coalesced b64 across the wave
                accA = wmma_f32x4(a1[ks], bf, accA);
                v2f b2 = w2[(size_t)ks * 32];
                accB = wmma_f32x4(a2[ks], b2, accB);
            }
        } else {
            #pragma unroll
            for (int ks = 0; ks + 1 < KS; ks += 2) {
                v2f bf0 = w1[(size_t)ks * 32];
                accA = wmma_f32x4(a1[ks], bf0, accA);
                v2f bf1 = w1[(size_t)(ks + 1) * 32];
                accB = wmma_f32x4(a1[ks + 1], bf1, accB);
            }
            if (KS & 1) {
                v2f bf = w1[(size_t)(KS - 1) * 32];
                accA = wmma_f32x4(a1[KS - 1], bf, accA);
            }
        }

        if (ncol < Nstore) {
            #pragma unroll
            for (int v = 0; v < 8; ++v) {
                const int row = rowBase + v + 8 * khalf;
                float val = accA[v] + accB[v];
                if (RELU) val = fmaxf(val, 0.0f);
                out[(size_t)row * Nstore + ncol] = val;
            }
        }
    }
}

// ---------------------------------------------------------------------------
// Pooling + head
// ---------------------------------------------------------------------------

__global__ void init_neg_inf_kernel(float* __restrict__ p, int n) {
    int i = blockIdx.x * blockDim.x + threadIdx.x;
    if (i < n) p[i] = -INFINITY;
}

__device__ inline void atomicMaxFloat(float* addr, float val) {
    if (val >= 0.0f) atomicMax((int*)addr, __float_as_int(val));
    else             atomicMin((unsigned int*)addr, __float_as_uint(val));
}

__global__ void segmax_kernel(const float* __restrict__ H, const int* __restrict__ batch,
                              float* __restrict__ G, int N, int F) {
    long long idx = (long long)blockIdx.x * blockDim.x + threadIdx.x;
    if (idx >= (long long)N * F) return;
    int node = (int)(idx / F);
    int f = (int)(idx - (long long)node * F);
    atomicMaxFloat(&G[(size_t)batch[node] * F + f], H[idx]);
}

__global__ void head_out_kernel(const float* __restrict__ G2, const float* __restrict__ Wo,
                                const float* __restrict__ bo, float* __restrict__ out,
                                int Gn, int F) {
    int g = blockIdx.x * blockDim.x + threadIdx.x;
    if (g >= Gn) return;
    float acc = bo[0];
    for (int k = 0; k < F; ++k) acc += G2[(size_t)g * F + k] * Wo[k];
    out[g] = acc;
}

// ---------------------------------------------------------------------------
// Host-side launch
// ---------------------------------------------------------------------------

static inline size_t align256(size_t x) { return (x + 255) & ~(size_t)255; }

extern "C" void kernel_launch(void* const* d_in, const int* in_sizes, int n_in,
                              void* d_out, int out_size, void* d_ws, size_t ws_size,
                              hipStream_t stream) {
    const int F1 = 35, F1P = 36, F2 = 128;
    const int N  = in_sizes[0] / F1;   // 100000
    const int E  = in_sizes[1] / 2;    // 3200000
    const int G  = out_size;           // 512

    const float* x     = (const float*)d_in[0];
    const int*   eidx  = (const int*)d_in[1];
    const int*   src   = eidx;
    const int*   dst   = eidx + E;
    const int*   batch = (const int*)d_in[2];
    const float* Wl1 = (const float*)d_in[3];
    const float* bl1 = (const float*)d_in[4];
    const float* Wr1 = (const float*)d_in[5];
    const float* Wl2 = (const float*)d_in[6];
    const float* bl2 = (const float*)d_in[7];
    const float* Wr2 = (const float*)d_in[8];
    const float* Wg1 = (const float*)d_in[9];
    const float* bg1 = (const float*)d_in[10];
    const float* Wg2 = (const float*)d_in[11];
    const float* bg2 = (const float*)d_in[12];
    const float* Wo  = (const float*)d_in[13];
    const float* bo  = (const float*)d_in[14];
    float* out = (float*)d_out;

    const int KS1 = F1P / 4;            // 9
    const int KS2 = F2 / 4;             // 32
    const int NT1 = 3;                  // ceil(35/16)
    const int NT2 = 8;                  // 128/16
    const size_t wf1sz = (size_t)NT1 * KS1 * 64;
    const size_t wf2sz = (size_t)NT2 * KS1 * 64;
    const size_t wfgsz = (size_t)NT2 * KS2 * 64;

    char* ws = (char*)d_ws;
    size_t off = 0;
    float* cnt  = (float*)(ws + off); off += align256((size_t)N * sizeof(float));
    float* xpad = (float*)(ws + off); off += align256((size_t)N * F1P * sizeof(float));
    float* sBuf = (float*)(ws + off); off += align256((size_t)N * F1P * sizeof(float));
    float* h1   = (float*)(ws + off); off += align256((size_t)N * F1P * sizeof(float));
    float* h2   = (float*)(ws + off); off += align256((size_t)N * F2 * sizeof(float));
    float* g0   = (float*)(ws + off); off += align256((size_t)G * F2 * sizeof(float));
    float* g1   = (float*)(ws + off); off += align256((size_t)G * F2 * sizeof(float));
    float* g2   = (float*)(ws + off); off += align256((size_t)G * F2 * sizeof(float));
    float* Wf1l = (float*)(ws + off); off += align256(wf1sz * sizeof(float));
    float* Wf1r = (float*)(ws + off); off += align256(wf1sz * sizeof(float));
    float* Wf2l = (float*)(ws + off); off += align256(wf2sz * sizeof(float));
    float* Wf2r = (float*)(ws + off); off += align256(wf2sz * sizeof(float));
    float* WfG1 = (float*)(ws + off); off += align256(wfgsz * sizeof(float));
    float* WfG2 = (float*)(ws + off); off += align256(wfgsz * sizeof(float));
    float* b1p  = (float*)(ws + off); off += align256((size_t)NT1 * 16 * sizeof(float));
    float* b2p  = (float*)(ws + off); off += align256((size_t)NT2 * 16 * sizeof(float));
    float* bg1p = (float*)(ws + off); off += align256((size_t)NT2 * 16 * sizeof(float));
    float* bg2p = (float*)(ws + off); off += align256((size_t)NT2 * 16 * sizeof(float));
    (void)ws_size;

    const int TB = 256;

    // ---- prep ----
    hipMemsetAsync(cnt, 0, (size_t)N * sizeof(float), stream);
    {
        long long tot = (long long)N * F1P;
        pad_rows_kernel<<<(int)((tot + TB - 1) / TB), TB, 0, stream>>>(x, xpad, N, F1, F1P);
    }
    pack_weights_kernel<<<(NT1 * KS1 * 32 + TB - 1) / TB, TB, 0, stream>>>(Wl1, bl1, F1, F1, NT1, KS1, Wf1l, b1p);
    pack_weights_kernel<<<(NT1 * KS1 * 32 + TB - 1) / TB, TB, 0, stream>>>(Wr1, bl1, F1, F1, NT1, KS1, Wf1r, b1p);
    pack_weights_kernel<<<(NT2 * KS1 * 32 + TB - 1) / TB, TB, 0, stream>>>(Wl2, bl2, F1, F2, NT2, KS1, Wf2l, b2p);
    pack_weights_kernel<<<(NT2 * KS1 * 32 + TB - 1) / TB, TB, 0, stream>>>(Wr2, bl2, F1, F2, NT2, KS1, Wf2r, b2p);
    pack_weights_kernel<<<(NT2 * KS2 * 32 + TB - 1) / TB, TB, 0, stream>>>(Wg1, bg1, F2, F2, NT2, KS2, WfG1, bg1p);
    pack_weights_kernel<<<(NT2 * KS2 * 32 + TB - 1) / TB, TB, 0, stream>>>(Wg2, bg2, F2, F2, NT2, KS2, WfG2, bg2p);

    // ---- degree / reciprocal ----
    degree_kernel<<<(E + TB - 1) / TB, TB, 0, stream>>>(dst, cnt, E);
    rcnt_kernel<<<(N + TB - 1) / TB, TB, 0, stream>>>(cnt, N);

    const int mGrid = (N / 16 + 3) / 4;
    const int gGrid = (G / 16 + 3) / 4;

    // ---- SAGE layer 1 ----
    hipMemsetAsync(sBuf, 0, (size_t)N * F1P * sizeof(float), stream);
    {
        long long tot = (long long)E * F1;
        scatter_sum_kernel<<<(int)((tot + TB - 1) / TB), TB, 0, stream>>>(xpad, src, dst, sBuf, E, F1, F1P, F1P);
    }
    fused_linear_wmma<F1P, true, true><<<mGrid, 128, 0, stream>>>(
        sBuf, cnt, Wf1l, xpad, Wf1r, b1p, h1, N, NT1, F1P);

    // ---- SAGE layer 2 ----
    hipMemsetAsync(sBuf, 0, (size_t)N * F1P * sizeof(float), stream);
    {
        long long tot = (long long)E * F1;
        scatter_sum_kernel<<<(int)((tot + TB - 1) / TB), TB, 0, stream>>>(h1, src, dst, sBuf, E, F1, F1P, F1P);
    }
    fused_linear_wmma<F1P, true, false><<<mGrid, 128, 0, stream>>>(
        sBuf, cnt, Wf2l, h1, Wf2r, b2p, h2, N, NT2, F2);

    // ---- global max pool ----
    init_neg_inf_kernel<<<(G * F2 + TB - 1) / TB, TB, 0, stream>>>(g0, G * F2);
    {
        long long tot = (long long)N * F2;
        segmax_kernel<<<(int)((tot + TB - 1) / TB), TB, 0, stream>>>(h2, batch, g0, N, F2);
    }

    // ---- FC head ----
    fused_linear_wmma<F2, false, true><<<gGrid, 128, 0, stream>>>(
        g0, nullptr, WfG1, nullptr, nullptr, bg1p, g1, G, NT2, F2);
    fused_linear_wmma<F2, false, true><<<gGrid, 128, 0, stream>>>(
        g1, nullptr, WfG2, nullptr, nullptr, bg2p, g2, G, NT2, F2);
    head_out_kernel<<<(G + TB - 1) / TB, TB, 0, stream>>>(g2, Wo, bo, out, G, F2);
}